// LogDomainNoiseSuppression_2619930050867
// MI455X (gfx1250) — compile-verified
//
#include <hip/hip_runtime.h>
#include <stdint.h>

// ---------------- problem constants (from reference) ----------------
#define CHN   5
#define BINS  256
#define NPC   (1u << 23)            // elements per reshaped channel = B*L = 8,388,608
#define RMAXF 8.0f
#define EPSF  1e-8f
#define ALPHAF 0.02f
#define THRESHF (-2.0f)

// ---------------- tiling ----------------
#define TPB 256
#define VPT 16                                   // float4 per thread
#define V4_PER_BLOCK (TPB * VPT)                 // 4096
#define V4_PER_CHAN  (NPC / 4)                   // 2,097,152
#define BLOCKS_PER_CHAN (V4_PER_CHAN / V4_PER_BLOCK)  // 512
#define GRID_BIG (CHN * BLOCKS_PER_CHAN)         // 2560 blocks

#define CAND_CAP (1u << 20)                      // candidate values per channel (~2.7x headroom)
#define RBLOCKS 64                               // refine blocks per channel
#define GRID_REF (CHN * RBLOCKS)

// ---------------- gfx1250 async global->LDS (guarded) ----------------
#if defined(__HIP_DEVICE_COMPILE__) && __has_builtin(__builtin_amdgcn_global_load_async_to_lds_b32)
#define HAVE_ASYNC_LDS 1
#else
#define HAVE_ASYNC_LDS 0
#endif

__device__ __forceinline__ void wait_async_zero() {
#if defined(__HIP_DEVICE_COMPILE__) && __has_builtin(__builtin_amdgcn_s_wait_asynccnt)
  __builtin_amdgcn_s_wait_asynccnt(0);
#else
  asm volatile("s_wait_asynccnt 0" ::: "memory");
#endif
}

// ---------------- kernels ----------------

// Re-initialize all device-side state (runs every kernel_launch call).
__global__ void k_init(uint32_t* __restrict__ counts, uint32_t* __restrict__ prefix,
                       uint32_t* __restrict__ ranks, uint32_t* __restrict__ new_hist,
                       uint32_t* __restrict__ cand_count) {
  for (int j = threadIdx.x; j < CHN * 2 * BINS; j += blockDim.x) counts[j] = 0u;
  for (int j = threadIdx.x; j < CHN * BINS; j += blockDim.x) new_hist[j] = 0u;
  if (threadIdx.x < CHN) cand_count[threadIdx.x] = 0u;
  if (threadIdx.x < CHN * 2) {
    prefix[threadIdx.x] = 0u;
    double pos = 0.99 * (double)(NPC - 1);        // 8304720.93
    uint32_t k = (uint32_t)pos;
    ranks[threadIdx.x] = k + (threadIdx.x & 1);   // target0 -> k, target1 -> k+1
  }
}

// Pass 0: full-data 256-bin histogram of the top 8 bits of |x| bit patterns.
__global__ __launch_bounds__(TPB) void k_pass0(const float* __restrict__ x,
                                               uint32_t* __restrict__ counts) {
  __shared__ uint32_t lh[4][BINS];                // 4-way replicated vs. atomic serialization
  for (int j = threadIdx.x; j < 4 * BINS; j += TPB) ((uint32_t*)lh)[j] = 0u;
  const int chan = blockIdx.x / BLOCKS_PER_CHAN;
  __syncthreads();
  const int rep = threadIdx.x & 3;
  const float4* xv = (const float4*)x;
  const size_t base = (size_t)blockIdx.x * V4_PER_BLOCK + threadIdx.x;
  for (int j = 0; j < VPT; ++j) {
    if (j + 2 < VPT) __builtin_prefetch(&xv[base + (size_t)(j + 2) * TPB], 0, 0);
    float4 v = xv[base + (size_t)j * TPB];
    uint32_t u[4];
    u[0] = __float_as_uint(v.x) & 0x7fffffffu;
    u[1] = __float_as_uint(v.y) & 0x7fffffffu;
    u[2] = __float_as_uint(v.z) & 0x7fffffffu;
    u[3] = __float_as_uint(v.w) & 0x7fffffffu;
#pragma unroll
    for (int e = 0; e < 4; ++e) atomicAdd(&lh[rep][u[e] >> 24], 1u);
  }
  __syncthreads();
  for (int b = threadIdx.x; b < BINS; b += TPB) {
    uint32_t s = lh[0][b] + lh[1][b] + lh[2][b] + lh[3][b];
    if (s) atomicAdd(&counts[(size_t)chan * 2 * BINS + b], s);   // target-0 slot
  }
}

// Tiny single-block scan: pick the digit bucket containing each rank, update
// prefix/rank state, and on the last pass produce the interpolated quantile.
__global__ void k_radix_scan(uint32_t* __restrict__ counts, uint32_t* __restrict__ prefix,
                             uint32_t* __restrict__ ranks, uint32_t* __restrict__ results,
                             float* __restrict__ max_vals, int pass) {
  const int c = threadIdx.x;
  if (c < CHN) {
    const uint32_t p0 = prefix[c * 2 + 0];
    const uint32_t p1 = prefix[c * 2 + 1];
    const bool dup = (pass == 0) || (p0 == p1);
    for (int t = 0; t < 2; ++t) {
      const uint32_t* cnt = &counts[((size_t)c * 2 + (dup ? 0 : t)) * BINS];
      uint32_t rr = ranks[c * 2 + t];
      uint32_t cum = 0, dig = 255u;
      for (int b = 0; b < BINS; ++b) {
        uint32_t cb = cnt[b];
        if (rr < cum + cb) { dig = (uint32_t)b; rr -= cum; break; }
        cum += cb;
      }
      ranks[c * 2 + t] = rr;
      uint32_t pr = ((t == 0 ? p0 : p1) << 8) | dig;
      prefix[c * 2 + t] = pr;
      if (pass == 3) results[c * 2 + t] = pr;
    }
    if (pass == 3) {
      float vk  = __uint_as_float(results[c * 2 + 0]);   // k-th order stat of |x|
      float vk1 = __uint_as_float(results[c * 2 + 1]);   // (k+1)-th order stat
      double pos  = 0.99 * (double)(NPC - 1);
      double frac = pos - (double)(uint32_t)pos;
      float q = (float)((double)vk + frac * ((double)vk1 - (double)vk));
      max_vals[c] = fmaxf(q, EPSF);
    }
  }
  __syncthreads();
  for (int j = threadIdx.x; j < CHN * 2 * BINS; j += blockDim.x) counts[j] = 0u;  // for next pass
}

// Compact all |x| bit patterns whose top-8 bits match either rank's bucket.
// Wave-aggregated atomic append (ballot + popc + one atomic per wave).
__global__ __launch_bounds__(TPB) void k_compact(const float* __restrict__ x,
                                                 const uint32_t* __restrict__ prefix,
                                                 uint32_t* __restrict__ cand,
                                                 uint32_t* __restrict__ cand_count) {
  const int chan = blockIdx.x / BLOCKS_PER_CHAN;
  const uint32_t p0 = prefix[chan * 2 + 0];
  const uint32_t p1 = prefix[chan * 2 + 1];
  uint32_t* cbuf = cand + (size_t)chan * CAND_CAP;
  const float4* xv = (const float4*)x;
  const size_t base = (size_t)blockIdx.x * V4_PER_BLOCK + threadIdx.x;
  const int lane = threadIdx.x & 31;
  for (int j = 0; j < VPT; ++j) {
    if (j + 2 < VPT) __builtin_prefetch(&xv[base + (size_t)(j + 2) * TPB], 0, 0);
    float4 v = xv[base + (size_t)j * TPB];
    uint32_t u[4];
    u[0] = __float_as_uint(v.x) & 0x7fffffffu;
    u[1] = __float_as_uint(v.y) & 0x7fffffffu;
    u[2] = __float_as_uint(v.z) & 0x7fffffffu;
    u[3] = __float_as_uint(v.w) & 0x7fffffffu;
#pragma unroll
    for (int e = 0; e < 4; ++e) {
      uint32_t hb = u[e] >> 24;
      bool match = (hb == p0) || (hb == p1);
      uint32_t mlo = (uint32_t)__ballot(match);
      if (mlo) {
        int leader = __ffs(mlo) - 1;
        uint32_t wbase = 0u;
        if (lane == leader) wbase = atomicAdd(&cand_count[chan], (uint32_t)__popc(mlo));
        wbase = (uint32_t)__shfl((int)wbase, leader, 32);
        if (match) {
          uint32_t pos = wbase + (uint32_t)__popc(mlo & ((1u << lane) - 1u));
          if (pos < CAND_CAP) cbuf[pos] = u[e];
        }
      }
    }
  }
}

// Refinement digit passes (1..3) over the compacted candidate stream only.
__global__ __launch_bounds__(TPB) void k_refine(const uint32_t* __restrict__ cand,
                                                const uint32_t* __restrict__ cand_count,
                                                const uint32_t* __restrict__ prefix,
                                                uint32_t* __restrict__ counts, int pass) {
  __shared__ uint32_t lh[4][2][BINS];
  for (int j = threadIdx.x; j < 4 * 2 * BINS; j += TPB) ((uint32_t*)lh)[j] = 0u;
  const int chan = blockIdx.x / RBLOCKS;
  const int blk  = blockIdx.x % RBLOCKS;
  const uint32_t p0 = prefix[chan * 2 + 0];
  const uint32_t p1 = prefix[chan * 2 + 1];
  const bool dup = (p0 == p1);
  const int shiftD = 24 - 8 * pass;
  const int shiftP = 32 - 8 * pass;               // pass >= 1 here, shift valid
  uint32_t n = cand_count[chan];
  if (n > CAND_CAP) n = CAND_CAP;
  __syncthreads();
  const int rep = threadIdx.x & 3;
  const uint32_t* cbuf = cand + (size_t)chan * CAND_CAP;
  for (uint32_t i = (uint32_t)blk * TPB + threadIdx.x; i < n; i += (uint32_t)RBLOCKS * TPB) {
    uint32_t u = cbuf[i];
    uint32_t dig = (u >> shiftD) & 0xFFu;
    uint32_t hp = u >> shiftP;
    if (dup) {
      if (hp == p0) atomicAdd(&lh[rep][0][dig], 1u);
    } else {
      if (hp == p0) atomicAdd(&lh[rep][0][dig], 1u);
      if (hp == p1) atomicAdd(&lh[rep][1][dig], 1u);
    }
  }
  __syncthreads();
  for (int j = threadIdx.x; j < 2 * BINS; j += TPB) {
    int t = j >> 8, b = j & 255;
    uint32_t s = lh[0][t][b] + lh[1][t][b] + lh[2][t][b] + lh[3][t][b];
    if (s) atomicAdd(&counts[((size_t)chan * 2 + t) * BINS + b], s);
  }
}

// 256-bin histogram of the normalized bin indices (exact integer counts).
__global__ __launch_bounds__(TPB) void k_hist(const float* __restrict__ x,
                                              const float* __restrict__ max_vals,
                                              uint32_t* __restrict__ new_hist) {
  __shared__ uint32_t lh[4][BINS];
  for (int j = threadIdx.x; j < 4 * BINS; j += TPB) ((uint32_t*)lh)[j] = 0u;
  const int chan = blockIdx.x / BLOCKS_PER_CHAN;
  const float mv = max_vals[chan];
  __syncthreads();
  const int rep = threadIdx.x & 3;
  const float4* xv = (const float4*)x;
  const size_t base = (size_t)blockIdx.x * V4_PER_BLOCK + threadIdx.x;
  for (int j = 0; j < VPT; ++j) {
    if (j + 2 < VPT) __builtin_prefetch(&xv[base + (size_t)(j + 2) * TPB], 0, 0);
    float4 v = xv[base + (size_t)j * TPB];
    float f[4] = {v.x, v.y, v.z, v.w};
#pragma unroll
    for (int e = 0; e < 4; ++e) {
      float a = fabsf(f[e]);
      float norm = a / mv * RMAXF;
      float clipped = fminf(norm, RMAXF - EPSF);
      int id = (int)(clipped * ((float)(BINS - 1) / RMAXF));
      id = id > (BINS - 1) ? (BINS - 1) : id;
      atomicAdd(&lh[rep][id], 1u);
    }
  }
  __syncthreads();
  for (int b = threadIdx.x; b < BINS; b += TPB) {
    uint32_t s = lh[0][b] + lh[1][b] + lh[2][b] + lh[3][b];
    if (s) atomicAdd(&new_hist[chan * BINS + b], s);
  }
}

// EMA + logp_obs + mask table: mask depends only on (channel, bin).
__global__ void k_table(const float* __restrict__ hist_in, const float* __restrict__ logp_ref,
                        const uint32_t* __restrict__ new_hist, float* __restrict__ mask_table) {
  __shared__ float red[BINS];
  const int b = threadIdx.x;   // 256 threads
  for (int c = 0; c < CHN; ++c) {
    float cnt = (float)new_hist[c * BINS + b];
    float h = (1.0f - ALPHAF) * hist_in[c * BINS + b] + ALPHAF * cnt;
    float sm = h + EPSF;
    red[b] = sm;
    __syncthreads();
    for (int s = BINS / 2; s > 0; s >>= 1) {
      if (b < s) red[b] += red[b + s];
      __syncthreads();
    }
    float sum = red[0];
    __syncthreads();
    float logp_obs = logf(sm / sum);
    float Lam = logp_ref[c * BINS + b] - logp_obs;
    float z = Lam - THRESHF;                         // mask = sigmoid(-z)
    mask_table[c * BINS + b] = 1.0f / (1.0f + expf(z));
  }
}

// Final streaming pass: out = x * mask_table[chan][idx(x)].
// Table is staged into LDS with gfx1250 async global->LDS loads.
__global__ __launch_bounds__(TPB) void k_apply(const float* __restrict__ x,
                                               const float* __restrict__ max_vals,
                                               const float* __restrict__ mask_table,
                                               float* __restrict__ out) {
  __shared__ float mt[BINS];
  const int chan = blockIdx.x / BLOCKS_PER_CHAN;
#if HAVE_ASYNC_LDS
  {
    void* gsrc = (void*)(mask_table + chan * BINS + threadIdx.x);   // one dword per lane
    __builtin_amdgcn_global_load_async_to_lds_b32(
        (__attribute__((address_space(1))) int*)gsrc,
        (__attribute__((address_space(3))) int*)&mt[threadIdx.x],
        0, 0);
    wait_async_zero();
  }
#else
  mt[threadIdx.x] = mask_table[chan * BINS + threadIdx.x];
#endif
  const float mv = max_vals[chan];
  __syncthreads();
  const float4* xv = (const float4*)x;
  float4* ov = (float4*)out;
  const size_t base = (size_t)blockIdx.x * V4_PER_BLOCK + threadIdx.x;
  for (int j = 0; j < VPT; ++j) {
    if (j + 2 < VPT) __builtin_prefetch(&xv[base + (size_t)(j + 2) * TPB], 0, 0);
    float4 v = xv[base + (size_t)j * TPB];
    float f[4] = {v.x, v.y, v.z, v.w};
    float g[4];
#pragma unroll
    for (int e = 0; e < 4; ++e) {
      float a = fabsf(f[e]);
      float norm = a / mv * RMAXF;
      float clipped = fminf(norm, RMAXF - EPSF);
      int id = (int)(clipped * ((float)(BINS - 1) / RMAXF));
      id = id > (BINS - 1) ? (BINS - 1) : id;
      g[e] = f[e] * mt[id];                          // ds_load_b32 gather
    }
    float4 r; r.x = g[0]; r.y = g[1]; r.z = g[2]; r.w = g[3];
    ov[base + (size_t)j * TPB] = r;
  }
}

// ---------------- launch ----------------
extern "C" void kernel_launch(void* const* d_in, const int* in_sizes, int n_in,
                              void* d_out, int out_size, void* d_ws, size_t ws_size,
                              hipStream_t stream) {
  (void)in_sizes; (void)n_in; (void)out_size; (void)ws_size;
  const float* x        = (const float*)d_in[0];   // (4,5,2097152) f32
  const float* hist_in  = (const float*)d_in[1];   // (5,256) f32
  const float* logp_ref = (const float*)d_in[2];   // (5,256) f32
  float* out = (float*)d_out;

  char* ws = (char*)d_ws;
  uint32_t* counts     = (uint32_t*)(ws + 0);          // 5*2*256 u32 = 10240 B
  uint32_t* prefix     = (uint32_t*)(ws + 10240);      // 10 u32
  uint32_t* ranks      = (uint32_t*)(ws + 10304);      // 10 u32
  uint32_t* results    = (uint32_t*)(ws + 10368);      // 10 u32
  float*    max_vals   = (float*)   (ws + 10432);      // 5 f32
  uint32_t* cand_count = (uint32_t*)(ws + 10496);      // 5 u32
  uint32_t* new_hist   = (uint32_t*)(ws + 10560);      // 5*256 u32 = 5120 B
  float*    mask_tab   = (float*)   (ws + 15680);      // 5*256 f32 = 5120 B
  uint32_t* cand       = (uint32_t*)(ws + 20992);      // 5 * 2^20 u32 = ~20 MB

  k_init<<<1, 256, 0, stream>>>(counts, prefix, ranks, new_hist, cand_count);
  k_pass0<<<GRID_BIG, TPB, 0, stream>>>(x, counts);
  k_radix_scan<<<1, 256, 0, stream>>>(counts, prefix, ranks, results, max_vals, 0);
  k_compact<<<GRID_BIG, TPB, 0, stream>>>(x, prefix, cand, cand_count);
  for (int p = 1; p < 4; ++p) {
    k_refine<<<GRID_REF, TPB, 0, stream>>>(cand, cand_count, prefix, counts, p);
    k_radix_scan<<<1, 256, 0, stream>>>(counts, prefix, ranks, results, max_vals, p);
  }
  k_hist<<<GRID_BIG, TPB, 0, stream>>>(x, max_vals, new_hist);
  k_table<<<1, BINS, 0, stream>>>(hist_in, logp_ref, new_hist, mask_tab);
  k_apply<<<GRID_BIG, TPB, 0, stream>>>(x, max_vals, mask_tab, out);
}